// PointReplacer_40166534152222
// MI455X (gfx1250) — compile-verified
//
#include <hip/hip_runtime.h>
#include <math.h>

// Problem constants (match reference)
#define BLV   64      // B*L
#define NPTS  1024    // N
#define MEMV  64      // MEM
#define TOK   1088    // N + MEM
#define DIMV  128     // DIM

typedef __attribute__((ext_vector_type(2))) float v2f;
typedef __attribute__((ext_vector_type(8))) float v8f;

__device__ inline v8f v8zero() {
    v8f z = {0.f,0.f,0.f,0.f,0.f,0.f,0.f,0.f};
    return z;
}

__device__ inline float gelu_exact(float x) {
    return 0.5f * x * (1.0f + erff(x * 0.70710678118654752f));
}

// C(16x16) += A(16xK, row-major lda) @ B(Kx16, row-major ldb), fp32 WMMA, K%4==0
__device__ inline v8f wmma16x16(const float* A, int lda, const float* B, int ldb,
                                int K, v8f acc) {
    const int lane = threadIdx.x & 31;
    const int rc   = lane & 15;       // A row / B col for this lane
    const int hi   = lane >> 4;       // half selector
    for (int k = 0; k < K; k += 4) {
        v2f a, b;
        a.x = A[rc * lda + k + 2*hi + 0];
        a.y = A[rc * lda + k + 2*hi + 1];
        b.x = B[(k + 2*hi + 0) * ldb + rc];
        b.y = B[(k + 2*hi + 1) * ldb + rc];
        acc = __builtin_amdgcn_wmma_f32_16x16x4_f32(false, a, false, b,
                                                    (short)0, acc, false, false);
    }
    return acc;
}

// C(16x16) += A(16xK, row-major lda) @ Bt(16xK, row-major ldb)^T  (i.e. A @ B^T)
__device__ inline v8f wmma16x16_bt(const float* A, int lda, const float* Bt, int ldb,
                                   int K, v8f acc) {
    const int lane = threadIdx.x & 31;
    const int rc   = lane & 15;
    const int hi   = lane >> 4;
    for (int k = 0; k < K; k += 4) {
        v2f a, b;
        a.x = A[rc * lda + k + 2*hi + 0];
        a.y = A[rc * lda + k + 2*hi + 1];
        b.x = Bt[rc * ldb + k + 2*hi + 0];
        b.y = Bt[rc * ldb + k + 2*hi + 1];
        acc = __builtin_amdgcn_wmma_f32_16x16x4_f32(false, a, false, b,
                                                    (short)0, acc, false, false);
    }
    return acc;
}

// ---------------------------------------------------------------------------
// K0: broadcast memory bank keys into xm tail:  xm[bl][N+m][d] = mem[0][d][m]
__global__ void k_memfill(const float* __restrict__ mem, float* __restrict__ xm) {
    int bl = blockIdx.x;
    for (int idx = threadIdx.x; idx < MEMV * DIMV; idx += blockDim.x) {
        int m = idx / DIMV, d = idx % DIMV;
        xm[((size_t)bl * TOK + NPTS + m) * DIMV + d] = mem[d * MEMV + m];
    }
}

// ---------------------------------------------------------------------------
// K1: encoder  pts(3) -> Linear -> LN -> GELU -> Linear(128x128 via WMMA)
__global__ __launch_bounds__(256) void k_encoder(
    const float* __restrict__ pts,
    const float* __restrict__ w1, const float* __restrict__ b1,
    const float* __restrict__ lng, const float* __restrict__ lnb,
    const float* __restrict__ w2, const float* __restrict__ b2,
    float* __restrict__ xm) {
    __shared__ float Asm[128 * 132];          // padded stride vs 64 banks
    const int bl = blockIdx.x;
    const int r0 = blockIdx.y * 128;          // 8 row-blocks of 128
    const int tid = threadIdx.x;

    // Linear(3->128)
    for (int idx = tid; idx < 128 * 128; idx += 256) {
        int r = idx >> 7, d = idx & 127;
        const float* p = pts + ((size_t)bl * NPTS + r0 + r) * 3;
        float h = b1[d] + p[0]*w1[0*DIMV+d] + p[1]*w1[1*DIMV+d] + p[2]*w1[2*DIMV+d];
        Asm[r * 132 + d] = h;
    }
    __syncthreads();

    // LayerNorm + exact GELU per row
    if (tid < 128) {
        float* row = &Asm[tid * 132];
        float mu = 0.f;
        for (int d = 0; d < 128; ++d) mu += row[d];
        mu *= (1.f / 128.f);
        float var = 0.f;
        for (int d = 0; d < 128; ++d) { float t = row[d] - mu; var += t * t; }
        var *= (1.f / 128.f);
        float rstd = rsqrtf(var + 1e-5f);
        for (int d = 0; d < 128; ++d) {
            float t = (row[d] - mu) * rstd * lng[d] + lnb[d];
            row[d] = gelu_exact(t);
        }
    }
    __syncthreads();

    // feat = A @ w2 + b2  (128x128 @ 128x128), 64 tiles over 8 waves
    const int wave = tid >> 5, lane = tid & 31;
    const int col = lane & 15, hi = lane >> 4;
    for (int t = wave; t < 64; t += 8) {
        int tr = (t >> 3) * 16, tc = (t & 7) * 16;
        v8f acc = v8zero();
        acc = wmma16x16(&Asm[tr * 132], 132, &w2[tc], DIMV, 128, acc);
        float bias = b2[tc + col];
        float* outp = xm + ((size_t)bl * TOK + r0 + tr + hi * 8) * DIMV + tc + col;
        #pragma unroll
        for (int vv = 0; vv < 8; ++vv) outp[(size_t)vv * DIMV] = acc[vv] + bias;
    }
}

// ---------------------------------------------------------------------------
// K2: qkv = xm @ qkv_w  ([64x128] @ [128x384] per block), split into q/k/v
__global__ __launch_bounds__(256) void k_qkv(
    const float* __restrict__ xm, const float* __restrict__ qkv_w,
    float* __restrict__ q, float* __restrict__ kk, float* __restrict__ v) {
    __shared__ float Asm[64 * 132];
    int bl = blockIdx.x, r0 = blockIdx.y * 64;
    int tid = threadIdx.x;
    const float* src = xm + ((size_t)bl * TOK + r0) * DIMV;
    for (int idx = tid; idx < 64 * 128; idx += 256) {
        int r = idx >> 7, d = idx & 127;
        Asm[r * 132 + d] = src[(size_t)r * DIMV + d];
    }
    __syncthreads();
    int wave = tid >> 5, lane = tid & 31, col = lane & 15, hi = lane >> 4;
    for (int t = wave; t < 96; t += 8) {            // 4 row-tiles x 24 col-tiles
        int tr = (t / 24) * 16, tc = (t % 24) * 16;
        v8f acc = v8zero();
        acc = wmma16x16(&Asm[tr * 132], 132, &qkv_w[tc], 3 * DIMV, 128, acc);
        int c = tc + col;
        float* dst;
        if (c < DIMV)            dst = q  + ((size_t)bl*TOK + r0 + tr + hi*8)*DIMV + c;
        else if (c < 2 * DIMV)   dst = kk + ((size_t)bl*TOK + r0 + tr + hi*8)*DIMV + (c - DIMV);
        else                     dst = v  + ((size_t)bl*TOK + r0 + tr + hi*8)*DIMV + (c - 2*DIMV);
        #pragma unroll
        for (int vv = 0; vv < 8; ++vv) dst[(size_t)vv * DIMV] = acc[vv];
    }
}

// ---------------------------------------------------------------------------
// K3: per-column softmax stats over i:  c_j = max_i <k_i,q_j>/s,  Z_j = sum exp(..-c_j)
__global__ __launch_bounds__(128) void k_colstats(
    const float* __restrict__ q, const float* __restrict__ kk,
    float* __restrict__ cmax, float* __restrict__ csum) {
    int bl = blockIdx.x;
    int jt = blockIdx.y * 4 + (threadIdx.x >> 5);   // 68 column-tiles
    int lane = threadIdx.x & 31, col = lane & 15, hi = lane >> 4;
    int j0 = jt * 16;
    const float scale = 0.08838834764831845f;       // 1/sqrt(128)
    const float* qrow = q + ((size_t)bl * TOK + j0) * DIMV;
    float runm = -INFINITY, runs = 0.f;
    for (int it = 0; it < TOK / 16; ++it) {
        const float* krow = kk + ((size_t)bl * TOK + it * 16) * DIMV;
        v8f s = v8zero();
        s = wmma16x16_bt(krow, DIMV, qrow, DIMV, 128, s);   // s[m][n] = k_i . q_j
        float bm = -INFINITY;
        #pragma unroll
        for (int t = 0; t < 8; ++t) { float x = s[t] * scale; s[t] = x; bm = fmaxf(bm, x); }
        bm = fmaxf(bm, __shfl_xor(bm, 16, 32));     // fold lane halves (same column)
        float bs = 0.f;
        #pragma unroll
        for (int t = 0; t < 8; ++t) bs += expf(s[t] - bm);
        bs += __shfl_xor(bs, 16, 32);
        float nm = fmaxf(runm, bm);
        runs = runs * expf(runm - nm) + bs * expf(bm - nm);
        runm = nm;
    }
    if (hi == 0) {
        cmax[(size_t)bl * TOK + j0 + col] = runm;
        csum[(size_t)bl * TOK + j0 + col] = runs;
    }
}

// ---------------------------------------------------------------------------
// K4: feat_mem[i,:] = sum_j exp(<k_i,q_j>/s - c_j)/Z_j * v[j,:]
__global__ __launch_bounds__(128) void k_attn(
    const float* __restrict__ q, const float* __restrict__ kk,
    const float* __restrict__ v,
    const float* __restrict__ cmax, const float* __restrict__ csum,
    float* __restrict__ fm) {
    __shared__ float Psm[4][16 * 20];               // per-wave padded P tile
    int bl = blockIdx.x;
    int wave = threadIdx.x >> 5;
    int it = blockIdx.y * 4 + wave;
    int lane = threadIdx.x & 31, col = lane & 15, hi = lane >> 4;
    int i0 = it * 16;
    const float scale = 0.08838834764831845f;
    const float* krow = kk + ((size_t)bl * TOK + i0) * DIMV;
    v8f out[8];
    #pragma unroll
    for (int c = 0; c < 8; ++c) out[c] = v8zero();
    float* P = Psm[wave];
    for (int jt = 0; jt < TOK / 16; ++jt) {
        int j0 = jt * 16;
        const float* qrow = q + ((size_t)bl * TOK + j0) * DIMV;
        v8f s = v8zero();
        s = wmma16x16_bt(krow, DIMV, qrow, DIMV, 128, s);
        float cj = cmax[(size_t)bl * TOK + j0 + col];
        float rz = 1.0f / csum[(size_t)bl * TOK + j0 + col];
        #pragma unroll
        for (int t = 0; t < 8; ++t)
            P[(t + 8 * hi) * 20 + col] = expf(s[t] * scale - cj) * rz;
        // wave-private LDS region; DS ops are in-order within a wave
        const float* vrow = v + ((size_t)bl * TOK + j0) * DIMV;
        #pragma unroll
        for (int c = 0; c < 8; ++c)
            out[c] = wmma16x16(P, 20, &vrow[c * 16], DIMV, 16, out[c]);
    }
    float* dst = fm + ((size_t)bl * TOK + i0 + hi * 8) * DIMV;
    #pragma unroll
    for (int c = 0; c < 8; ++c)
        #pragma unroll
        for (int t = 0; t < 8; ++t)
            dst[(size_t)t * DIMV + c * 16 + col] = out[c][t];
}

// ---------------------------------------------------------------------------
// K5: two heads: LN -> Linear(128x128, WMMA) -> GELU -> tiny Linear (1 / 3)
__global__ __launch_bounds__(256) void k_heads(
    const float* __restrict__ fm,
    const float* __restrict__ cf_ln_g, const float* __restrict__ cf_ln_b,
    const float* __restrict__ cf_w1,  const float* __restrict__ cf_b1,
    const float* __restrict__ cf_w2,  const float* __restrict__ cf_b2,
    const float* __restrict__ pc_ln_g, const float* __restrict__ pc_ln_b,
    const float* __restrict__ pc_w1,  const float* __restrict__ pc_b1,
    const float* __restrict__ pc_w2,  const float* __restrict__ pc_b2,
    float* __restrict__ conf, float* __restrict__ prec) {
    __shared__ float Ain[64 * 132];
    __shared__ float Hout[64 * 132];
    int bl = blockIdx.x, r0 = blockIdx.y * 64;
    int tid = threadIdx.x;
    int wave = tid >> 5, lane = tid & 31, col = lane & 15, hi = lane >> 4;

    for (int head = 0; head < 2; ++head) {
        const float* lng  = head ? pc_ln_g : cf_ln_g;
        const float* lnbp = head ? pc_ln_b : cf_ln_b;
        const float* w1   = head ? pc_w1   : cf_w1;
        const float* b1   = head ? pc_b1   : cf_b1;
        if (tid < 64) {
            const float* src = fm + ((size_t)bl * TOK + r0 + tid) * DIMV;
            float mu = 0.f;
            for (int d = 0; d < 128; ++d) mu += src[d];
            mu *= (1.f / 128.f);
            float var = 0.f;
            for (int d = 0; d < 128; ++d) { float t = src[d] - mu; var += t * t; }
            var *= (1.f / 128.f);
            float rstd = rsqrtf(var + 1e-5f);
            for (int d = 0; d < 128; ++d)
                Ain[tid * 132 + d] = (src[d] - mu) * rstd * lng[d] + lnbp[d];
        }
        __syncthreads();
        for (int t = wave; t < 32; t += 8) {        // 4 row-tiles x 8 col-tiles
            int tr = (t >> 3) * 16, tc = (t & 7) * 16;
            v8f acc = v8zero();
            acc = wmma16x16(&Ain[tr * 132], 132, &w1[tc], DIMV, 128, acc);
            float bias = b1[tc + col];
            #pragma unroll
            for (int vv = 0; vv < 8; ++vv)
                Hout[(tr + vv + 8 * hi) * 132 + tc + col] = gelu_exact(acc[vv] + bias);
        }
        __syncthreads();
        if (head == 0) {
            if (tid < 64) {
                float s = cf_b2[0];
                for (int d = 0; d < 128; ++d) s += Hout[tid * 132 + d] * cf_w2[d];
                conf[(size_t)bl * TOK + r0 + tid] = s;
            }
        } else {
            if (tid < 192) {
                int r = tid / 3, c = tid % 3;
                float s = pc_b2[c];
                for (int d = 0; d < 128; ++d) s += Hout[r * 132 + d] * pc_w2[d * 3 + c];
                prec[((size_t)bl * TOK + r0 + r) * 3 + c] = s;
            }
        }
        __syncthreads();
    }
}

// ---------------------------------------------------------------------------
// K6/K7: deterministic two-stage reduction for l_prec
__global__ __launch_bounds__(256) void k_prec_partial(
    const float* __restrict__ pts, const float* __restrict__ prec,
    float* __restrict__ part) {
    __shared__ float red[256];
    int bl = blockIdx.x, tid = threadIdx.x;
    const float* p = pts + (size_t)bl * NPTS * 3;
    const float* r = prec + (size_t)bl * TOK * 3;   // first N rows align with pts
    float s = 0.f;
    for (int i = tid; i < NPTS * 3; i += 256) {
        float d = r[i] - p[i];
        s += d * d;
    }
    red[tid] = s;
    __syncthreads();
    for (int off = 128; off > 0; off >>= 1) {
        if (tid < off) red[tid] += red[tid + off];
        __syncthreads();
    }
    if (tid == 0) part[bl] = red[0];
}

__global__ void k_prec_final(const float* __restrict__ part, float* __restrict__ out) {
    if (threadIdx.x == 0) {
        float s = 0.f;
        for (int i = 0; i < BLV; ++i) s += part[i];
        out[0] = s / (float)(BLV * NPTS * 3);
    }
}

// ---------------------------------------------------------------------------
// K8: conf[:, :N] -> output
__global__ void k_conf_out(const float* __restrict__ conf, float* __restrict__ out) {
    int bl = blockIdx.x;
    for (int i = threadIdx.x; i < NPTS; i += blockDim.x)
        out[(size_t)bl * NPTS + i] = conf[(size_t)bl * TOK + i];
}

// ---------------------------------------------------------------------------
// K9: stable rank-based bottom-M removal (replicates jax top_k tie-breaking)
__global__ __launch_bounds__(256) void k_mask_points(
    const float* __restrict__ conf, const float* __restrict__ pts,
    const float* __restrict__ prec, float* __restrict__ out) {
    __shared__ float c[TOK];
    int bl = blockIdx.x, tid = threadIdx.x;
    for (int i = tid; i < TOK; i += 256) c[i] = conf[(size_t)bl * TOK + i];
    __syncthreads();
    for (int t = tid; t < TOK; t += 256) {
        float ct = c[t];
        int rank = 0;
        for (int u = 0; u < TOK; ++u) {
            float cu = c[u];
            rank += (cu < ct) || (cu == ct && u < t);
        }
        float keep = (rank >= MEMV) ? 1.f : 0.f;
        const float* src = (t < NPTS) ? (pts + ((size_t)bl * NPTS + t) * 3)
                                      : (prec + ((size_t)bl * TOK + t) * 3);
        float* dst = out + ((size_t)bl * TOK + t) * 3;
        dst[0] = src[0] * keep;
        dst[1] = src[1] * keep;
        dst[2] = src[2] * keep;
    }
}

// ---------------------------------------------------------------------------
extern "C" void kernel_launch(void* const* d_in, const int* in_sizes, int n_in,
                              void* d_out, int out_size, void* d_ws, size_t ws_size,
                              hipStream_t stream) {
    const float* points  = (const float*)d_in[0];
    const float* enc_w1  = (const float*)d_in[1];
    const float* enc_b1  = (const float*)d_in[2];
    const float* enc_lng = (const float*)d_in[3];
    const float* enc_lnb = (const float*)d_in[4];
    const float* enc_w2  = (const float*)d_in[5];
    const float* enc_b2  = (const float*)d_in[6];
    const float* memb    = (const float*)d_in[7];
    const float* qkv_w   = (const float*)d_in[8];
    const float* pc_ln_g = (const float*)d_in[9];
    const float* pc_ln_b = (const float*)d_in[10];
    const float* pc_w1   = (const float*)d_in[11];
    const float* pc_b1   = (const float*)d_in[12];
    const float* pc_w2   = (const float*)d_in[13];
    const float* pc_b2   = (const float*)d_in[14];
    const float* cf_ln_g = (const float*)d_in[15];
    const float* cf_ln_b = (const float*)d_in[16];
    const float* cf_w1   = (const float*)d_in[17];
    const float* cf_b1   = (const float*)d_in[18];
    const float* cf_w2   = (const float*)d_in[19];
    const float* cf_b2   = (const float*)d_in[20];

    const size_t XMF = (size_t)BLV * TOK * DIMV;   // 8,912,896 floats
    float* ws    = (float*)d_ws;
    float* xm    = ws;                 // token features; reused as feat_mem
    float* q     = ws + XMF;
    float* kk    = ws + 2 * XMF;
    float* vv    = ws + 3 * XMF;
    float* cmaxp = ws + 4 * XMF;
    float* csump = cmaxp + (size_t)BLV * TOK;
    float* confp = csump + (size_t)BLV * TOK;
    float* precp = confp + (size_t)BLV * TOK;
    float* partp = precp + (size_t)BLV * TOK * 3;

    float* outp     = (float*)d_out;
    float* out_pts  = outp;                                  // [BL,TOK,3]
    float* out_conf = outp + (size_t)BLV * TOK * 3;          // [BL,N]
    float* out_lp   = out_conf + (size_t)BLV * NPTS;         // scalar

    k_memfill   <<<dim3(BLV),      256, 0, stream>>>(memb, xm);
    k_encoder   <<<dim3(BLV, 8),   256, 0, stream>>>(points, enc_w1, enc_b1,
                                                     enc_lng, enc_lnb, enc_w2, enc_b2, xm);
    k_qkv       <<<dim3(BLV, 17),  256, 0, stream>>>(xm, qkv_w, q, kk, vv);
    k_colstats  <<<dim3(BLV, 17),  128, 0, stream>>>(q, kk, cmaxp, csump);
    k_attn      <<<dim3(BLV, 17),  128, 0, stream>>>(q, kk, vv, cmaxp, csump, xm);
    k_heads     <<<dim3(BLV, 17),  256, 0, stream>>>(xm,
                                                     cf_ln_g, cf_ln_b, cf_w1, cf_b1, cf_w2, cf_b2,
                                                     pc_ln_g, pc_ln_b, pc_w1, pc_b1, pc_w2, pc_b2,
                                                     confp, precp);
    k_prec_partial<<<dim3(BLV),    256, 0, stream>>>(points, precp, partp);
    k_prec_final  <<<1,             32, 0, stream>>>(partp, out_lp);
    k_conf_out    <<<dim3(BLV),    256, 0, stream>>>(confp, out_conf);
    k_mask_points <<<dim3(BLV),    256, 0, stream>>>(confp, points, precp, out_pts);
}